// GCN_86397562126689
// MI455X (gfx1250) — compile-verified
//
#include <hip/hip_runtime.h>
#include <hip/hip_bf16.h>

#define USER_NUM 50000
#define ITEM_NUM 50000
#define NTOT     100000
#define VT_DIM   512
#define CAT_DIM  32
#define HIDDEN   128
#define OUT_DIM  64
#define EDGES    2000000
#define BAG      5
#define KFUSE    (VT_DIM + CAT_DIM)   // 544

// LDS row pads (floats): keep 16B row alignment, avoid 64-bank alignment
#define FPAD_F   548                  // fuse tile row stride  (548 % 64 = 36)
#define FPAD_H   132                  // sage tile row stride  (132 % 64 = 4)

typedef __attribute__((ext_vector_type(2))) float v2f;
typedef __attribute__((ext_vector_type(8))) float v8f;
typedef __attribute__((ext_vector_type(4))) unsigned int v4u;
typedef __attribute__((ext_vector_type(4))) int v4i;
typedef __attribute__((ext_vector_type(8))) int v8i;

// ---- CDNA5 async global->LDS copy (16B per lane), tracked with ASYNCcnt ----
__device__ __forceinline__ void async_copy_b128(const float* g, float* l) {
    auto lp = (__attribute__((address_space(3))) float*)l;
    asm volatile("global_load_async_to_lds_b128 %0, %1, off"
                 :: "v"(lp), "v"(g) : "memory");
}
__device__ __forceinline__ void async_wait0() {
    asm volatile("s_wait_asynccnt 0x0" ::: "memory");
}

__device__ __forceinline__ unsigned lds_offset_of(const void* p) {
    return (unsigned)(uintptr_t)(__attribute__((address_space(3))) const void*)p;
}

// ---- CDNA5 Tensor Data Mover: 2D tile global->LDS, HW row padding ----
// pad_interval code 6 = every 128 DWORDs, pad_amount code 3 = +4 DWORDs
// -> LDS row stride 132 floats for a 128-float tile row (bank-conflict free).
__device__ __forceinline__ void tdm_load_2d(const float* gbase, unsigned lds_off,
                                            unsigned tensor_d0, unsigned tensor_d1,
                                            unsigned tile_d0, unsigned tile_d1,
                                            unsigned stride0,
                                            unsigned pad_interval, unsigned pad_amount) {
    unsigned long long ga = (unsigned long long)(uintptr_t)gbase;
    v4u g0;
    g0.x = 1u;                                         // count=1 (valid), user mode
    g0.y = lds_off;                                    // lds_addr (bytes)
    g0.z = (unsigned)(ga & 0xFFFFFFFFu);               // global_addr[31:0]
    g0.w = (unsigned)((ga >> 32) & 0x1FFFFFFu)         // global_addr[56:32]
         | (2u << 30);                                 // type = 2 ("image")
    v8i g1;
    g1[0] = (int)((2u << 16)                           // data_size = 4B
                | (1u << 20)                           // pad_enable
                | (pad_interval << 22)
                | (pad_amount << 25));
    g1[1] = (int)(tensor_d0 << 16);                    // abar=0 | tensor_dim0[15:0]
    g1[2] = (int)((tensor_d0 >> 16) | (tensor_d1 << 16));
    g1[3] = (int)((tensor_d1 >> 16) | (tile_d0 << 16));
    g1[4] = (int)(tile_d1 & 0xFFFFu);                  // tile_dim1 | tile_dim2=0
    g1[5] = (int)stride0;                              // tensor_dim0_stride[31:0]
    g1[6] = 0;                                         // stride0 hi | stride1 lo
    g1[7] = 0;                                         // stride1 hi
    v4i gz = {0, 0, 0, 0};
#if __has_include(<hip/amd_detail/amd_gfx1250_TDM.h>)
    v8i gz8 = {0, 0, 0, 0, 0, 0, 0, 0};
    __builtin_amdgcn_tensor_load_to_lds(g0, g1, gz, gz, gz8, 0);
#else
    __builtin_amdgcn_tensor_load_to_lds(g0, g1, gz, gz, 0);
#endif
}

// ---------------- EmbeddingBag mean ----------------
__global__ void k_cat_emb(const float* __restrict__ table,
                          const int* __restrict__ idx,
                          const int* __restrict__ off,
                          float* __restrict__ cat_emb) {
    int t = blockIdx.x * blockDim.x + threadIdx.x;
    if (t >= ITEM_NUM * CAT_DIM) return;
    int item = t / CAT_DIM;
    int d    = t % CAT_DIM;
    int s = off[item];
    int e = (item + 1 < ITEM_NUM) ? off[item + 1] : ITEM_NUM * BAG;
    float sum = 0.0f;
    for (int j = s; j < e; ++j) sum += table[idx[j] * CAT_DIM + d];
    float cnt = (float)(e - s);
    cat_emb[t] = sum / fmaxf(cnt, 1.0f);
}

// ---------------- Fuse GEMM: x0[item] = [vt | cat] @ fuse_w^T ----------------
// block = 256 threads (8 waves); A tile (16 x 544) staged via ASYNC-to-LDS.
__global__ __launch_bounds__(256)
void k_fuse(const float* __restrict__ vt, const float* __restrict__ catE,
            const float* __restrict__ W, float* __restrict__ x0) {
    __shared__ alignas(16) float sA[16 * FPAD_F];

    int lane = threadIdx.x & 31;
    int wave = threadIdx.x >> 5;
    int half = lane >> 4;
    int l    = lane & 15;
    int m0   = blockIdx.x * 16;
    int col  = wave * 16 + l;     // output column / W row

    // stage vt rows: 16 x 512 floats = 2048 x b128 transfers
    const float* gv = vt + (size_t)m0 * VT_DIM;
#pragma unroll
    for (int r = 0; r < 8; ++r) {
        int t   = threadIdx.x + r * 256;        // 0..2047
        int row = t >> 7;
        int off = (t & 127) << 2;
        async_copy_b128(gv + (size_t)row * VT_DIM + off, &sA[row * FPAD_F + off]);
    }
    // stage cat rows: 16 x 32 floats = 128 x b128 transfers
    if (threadIdx.x < 128) {
        int row = threadIdx.x >> 3;
        int off = (threadIdx.x & 7) << 2;
        async_copy_b128(catE + (size_t)(m0 + row) * CAT_DIM + off,
                        &sA[row * FPAD_F + VT_DIM + off]);
    }
    async_wait0();
    __syncthreads();

    v8f c = {};
    for (int k0 = 0; k0 < KFUSE; k0 += 4) {
        int k = k0 + 2 * half;
        v2f a = *(const v2f*)&sA[l * FPAD_F + k];
        v2f b;
        b.x = W[(size_t)col * KFUSE + k];
        b.y = W[(size_t)col * KFUSE + k + 1];
        c = __builtin_amdgcn_wmma_f32_16x16x4_f32(false, a, false, b,
                                                  (short)0, c, false, false);
    }
#pragma unroll
    for (int v = 0; v < 8; ++v) {
        int r = m0 + v + 8 * half;
        x0[(size_t)(USER_NUM + r) * HIDDEN + col] = c[v];
    }
}

// ---------------- degree / reciprocal degree ----------------
__global__ void k_deg(const int* __restrict__ ei, float* __restrict__ deg) {
    int e = blockIdx.x * blockDim.x + threadIdx.x;
    if (e >= EDGES) return;
    atomicAdd(&deg[ei[EDGES + e]], 1.0f);
}

__global__ void k_rdeg(float* __restrict__ deg) {
    int i = blockIdx.x * blockDim.x + threadIdx.x;
    if (i >= NTOT) return;
    deg[i] = 1.0f / fmaxf(deg[i], 1.0f);
}

// ---------------- edge scatter: agg[dst] += x[src] (wave per edge) ----------------
__global__ __launch_bounds__(256)
void k_scatter(const float* __restrict__ x, const int* __restrict__ ei,
               float* __restrict__ agg) {
    int gw   = (blockIdx.x * blockDim.x + threadIdx.x) >> 5;   // edge id
    int lane = threadIdx.x & 31;
    int src = ei[gw];
    int dst = ei[EDGES + gw];
    const float4* xs = (const float4*)(x + (size_t)src * HIDDEN);
    float4 v = xs[lane];                       // 32 lanes * float4 = 128 floats
    float* ad = agg + (size_t)dst * HIDDEN + lane * 4;
    atomicAdd(ad + 0, v.x);
    atomicAdd(ad + 1, v.y);
    atomicAdd(ad + 2, v.z);
    atomicAdd(ad + 3, v.w);
}

// ---------------- SAGE layer 1: x1 = lrelu((agg/deg)@Wl^T + b + x0@Wr^T) -------------
// A tiles staged with the Tensor Data Mover (HW-padded to 132-float rows).
__global__ __launch_bounds__(256)
void k_sage1(const float* __restrict__ agg, const float* __restrict__ rdeg,
             const float* __restrict__ x0, const float* __restrict__ Wl,
             const float* __restrict__ b,  const float* __restrict__ Wr,
             float* __restrict__ x1) {
    __shared__ alignas(16) float sA[16 * FPAD_H];
    __shared__ alignas(16) float sX[16 * FPAD_H];

    int lane = threadIdx.x & 31;
    int wave = threadIdx.x >> 5;
    int half = lane >> 4;
    int l    = lane & 15;
    int m0   = blockIdx.x * 16;
    int col  = wave * 16 + l;

    if (threadIdx.x < 32) {   // wave 0 drives the TDM
        tdm_load_2d(agg + (size_t)m0 * HIDDEN, lds_offset_of(sA),
                    HIDDEN, NTOT, HIDDEN, 16, HIDDEN, /*pad*/6, 3);
        tdm_load_2d(x0  + (size_t)m0 * HIDDEN, lds_offset_of(sX),
                    HIDDEN, NTOT, HIDDEN, 16, HIDDEN, /*pad*/6, 3);
        __builtin_amdgcn_s_wait_tensorcnt(0);
    }
    __syncthreads();

    float rd = rdeg[m0 + l];

    v8f c = {};
    for (int k0 = 0; k0 < HIDDEN; k0 += 4) {
        int k = k0 + 2 * half;
        v2f a = *(const v2f*)&sA[l * FPAD_H + k];
        a.x *= rd; a.y *= rd;
        v2f bb;
        bb.x = Wl[col * HIDDEN + k];
        bb.y = Wl[col * HIDDEN + k + 1];
        c = __builtin_amdgcn_wmma_f32_16x16x4_f32(false, a, false, bb,
                                                  (short)0, c, false, false);
    }
    for (int k0 = 0; k0 < HIDDEN; k0 += 4) {
        int k = k0 + 2 * half;
        v2f a = *(const v2f*)&sX[l * FPAD_H + k];
        v2f bb;
        bb.x = Wr[col * HIDDEN + k];
        bb.y = Wr[col * HIDDEN + k + 1];
        c = __builtin_amdgcn_wmma_f32_16x16x4_f32(false, a, false, bb,
                                                  (short)0, c, false, false);
    }
    float bias = b[col];
#pragma unroll
    for (int v = 0; v < 8; ++v) {
        float val = c[v] + bias;
        val = (val > 0.0f) ? val : 0.01f * val;       // leaky_relu(0.01)
        int r = m0 + v + 8 * half;
        x1[(size_t)r * HIDDEN + col] = val;
    }
}

// ---------------- SAGE layer 2 (OUT=64, 4 waves/block, no relu) ----------------
__global__ __launch_bounds__(128)
void k_sage2(const float* __restrict__ agg, const float* __restrict__ rdeg,
             const float* __restrict__ x1, const float* __restrict__ Wl,
             const float* __restrict__ b,  const float* __restrict__ Wr,
             float* __restrict__ out) {
    __shared__ alignas(16) float sA[16 * FPAD_H];
    __shared__ alignas(16) float sX[16 * FPAD_H];

    int lane = threadIdx.x & 31;
    int wave = threadIdx.x >> 5;
    int half = lane >> 4;
    int l    = lane & 15;
    int m0   = blockIdx.x * 16;
    int col  = wave * 16 + l;      // 0..63

    if (threadIdx.x < 32) {   // wave 0 drives the TDM
        tdm_load_2d(agg + (size_t)m0 * HIDDEN, lds_offset_of(sA),
                    HIDDEN, NTOT, HIDDEN, 16, HIDDEN, /*pad*/6, 3);
        tdm_load_2d(x1  + (size_t)m0 * HIDDEN, lds_offset_of(sX),
                    HIDDEN, NTOT, HIDDEN, 16, HIDDEN, /*pad*/6, 3);
        __builtin_amdgcn_s_wait_tensorcnt(0);
    }
    __syncthreads();

    float rd = rdeg[m0 + l];

    v8f c = {};
    for (int k0 = 0; k0 < HIDDEN; k0 += 4) {
        int k = k0 + 2 * half;
        v2f a = *(const v2f*)&sA[l * FPAD_H + k];
        a.x *= rd; a.y *= rd;
        v2f bb;
        bb.x = Wl[col * HIDDEN + k];
        bb.y = Wl[col * HIDDEN + k + 1];
        c = __builtin_amdgcn_wmma_f32_16x16x4_f32(false, a, false, bb,
                                                  (short)0, c, false, false);
    }
    for (int k0 = 0; k0 < HIDDEN; k0 += 4) {
        int k = k0 + 2 * half;
        v2f a = *(const v2f*)&sX[l * FPAD_H + k];
        v2f bb;
        bb.x = Wr[col * HIDDEN + k];
        bb.y = Wr[col * HIDDEN + k + 1];
        c = __builtin_amdgcn_wmma_f32_16x16x4_f32(false, a, false, bb,
                                                  (short)0, c, false, false);
    }
    float bias = b[col];
#pragma unroll
    for (int v = 0; v < 8; ++v) {
        int r = m0 + v + 8 * half;
        out[(size_t)r * OUT_DIM + col] = c[v] + bias;
    }
}

extern "C" void kernel_launch(void* const* d_in, const int* in_sizes, int n_in,
                              void* d_out, int out_size, void* d_ws, size_t ws_size,
                              hipStream_t stream) {
    const float* vt      = (const float*)d_in[0];
    const int*   cat_idx = (const int*)  d_in[1];
    const int*   cat_off = (const int*)  d_in[2];
    const int*   ei      = (const int*)  d_in[3];
    const float* table   = (const float*)d_in[4];
    const float* fuse_w  = (const float*)d_in[5];
    const float* user    = (const float*)d_in[6];
    const float* w1_l    = (const float*)d_in[7];
    const float* b1      = (const float*)d_in[8];
    const float* w1_r    = (const float*)d_in[9];
    const float* w2_l    = (const float*)d_in[10];
    const float* b2      = (const float*)d_in[11];
    const float* w2_r    = (const float*)d_in[12];
    float* out = (float*)d_out;

    // workspace layout (floats)
    float* cat_emb = (float*)d_ws;                         // 50000*32
    float* x0      = cat_emb + (size_t)ITEM_NUM * CAT_DIM; // 100000*128
    float* agg     = x0  + (size_t)NTOT * HIDDEN;          // 100000*128
    float* x1      = agg + (size_t)NTOT * HIDDEN;          // 100000*128
    float* rdeg    = x1  + (size_t)NTOT * HIDDEN;          // 100000

    // zero accumulators
    hipMemsetAsync(agg,  0, (size_t)NTOT * HIDDEN * sizeof(float), stream);
    hipMemsetAsync(rdeg, 0, (size_t)NTOT * sizeof(float), stream);

    // 1) embedding bag mean
    k_cat_emb<<<(ITEM_NUM * CAT_DIM + 255) / 256, 256, 0, stream>>>(table, cat_idx, cat_off, cat_emb);

    // 2) fuse GEMM into item half of x0
    k_fuse<<<ITEM_NUM / 16, 256, 0, stream>>>(vt, cat_emb, fuse_w, x0);

    // 3) user rows into top of x0
    hipMemcpyAsync(x0, user, (size_t)USER_NUM * HIDDEN * sizeof(float),
                   hipMemcpyDeviceToDevice, stream);

    // 4) degree then reciprocal
    k_deg<<<(EDGES + 255) / 256, 256, 0, stream>>>(ei, rdeg);
    k_rdeg<<<(NTOT + 255) / 256, 256, 0, stream>>>(rdeg);

    // 5) layer-1 aggregation + SAGE conv
    k_scatter<<<EDGES / 8, 256, 0, stream>>>(x0, ei, agg);       // wave per edge
    k_sage1<<<NTOT / 16, 256, 0, stream>>>(agg, rdeg, x0, w1_l, b1, w1_r, x1);

    // 6) layer-2 aggregation + SAGE conv -> d_out
    hipMemsetAsync(agg, 0, (size_t)NTOT * HIDDEN * sizeof(float), stream);
    k_scatter<<<EDGES / 8, 256, 0, stream>>>(x1, ei, agg);
    k_sage2<<<NTOT / 16, 128, 0, stream>>>(agg, rdeg, x1, w2_l, b2, w2_r, out);
}